// QuantumEncoder_69939247448631
// MI455X (gfx1250) — compile-verified
//
#include <hip/hip_runtime.h>

typedef float v2f __attribute__((ext_vector_type(2)));
typedef float v8f __attribute__((ext_vector_type(8)));

#define NQ   4
#define DIM  16          // 2^NQ amplitudes
#define DINP 1024
#define LOUT 64
#define ROWS_PER_BLOCK 32
#define KSPLIT 4
#define KQ (DINP / KSPLIT)

// ---------------------------------------------------------------------------
// Kernel 1: build the shared 16x16 complex unitary V (2 layers of Rx/Ry/Rz per
// qubit + CX ring, from quantum_params). Lane k propagates basis column k.
// ---------------------------------------------------------------------------
__global__ void qe_build_unitary(const float* __restrict__ qp,
                                 float* __restrict__ Vr,
                                 float* __restrict__ Vi)
{
    const int col = threadIdx.x;
    if (col >= DIM) return;

    float sr[DIM], si[DIM];
#pragma unroll
    for (int b = 0; b < DIM; ++b) { sr[b] = (b == col) ? 1.0f : 0.0f; si[b] = 0.0f; }

    const int cxc[NQ] = {0, 1, 2, 3};
    const int cxt[NQ] = {1, 2, 3, 0};

#pragma unroll
    for (int layer = 0; layer < 2; ++layer) {
#pragma unroll
        for (int q = 0; q < NQ; ++q) {
            const int bit = 1 << q;
            const float th = qp[layer * NQ + q];
            const float c = cosf(0.5f * th);
            const float s = sinf(0.5f * th);
            // Rx = [[c, -i s], [-i s, c]]
#pragma unroll
            for (int i0 = 0; i0 < DIM; ++i0) {
                if (i0 & bit) continue;
                const int i1 = i0 | bit;
                float ar = sr[i0], ai = si[i0], br = sr[i1], bi = si[i1];
                sr[i0] = c*ar + s*bi;  si[i0] = c*ai - s*br;
                sr[i1] = c*br + s*ai;  si[i1] = c*bi - s*ar;
            }
            // Ry = [[c, -s], [s, c]]
#pragma unroll
            for (int i0 = 0; i0 < DIM; ++i0) {
                if (i0 & bit) continue;
                const int i1 = i0 | bit;
                float ar = sr[i0], ai = si[i0], br = sr[i1], bi = si[i1];
                sr[i0] = c*ar - s*br;  si[i0] = c*ai - s*bi;
                sr[i1] = s*ar + c*br;  si[i1] = s*ai + c*bi;
            }
            // Rz = diag(e^{-i th/2}, e^{+i th/2})
#pragma unroll
            for (int i0 = 0; i0 < DIM; ++i0) {
                if (i0 & bit) continue;
                const int i1 = i0 | bit;
                float ar = sr[i0], ai = si[i0], br = sr[i1], bi = si[i1];
                sr[i0] = c*ar + s*ai;  si[i0] = c*ai - s*ar;
                sr[i1] = c*br - s*bi;  si[i1] = c*bi + s*br;
            }
        }
        // CX chain: (0,1),(1,2),(2,3),(3,0)  — swap amp(b) <-> amp(b^tb) when ctrl bit set
#pragma unroll
        for (int e = 0; e < NQ; ++e) {
            const int cb = 1 << cxc[e], tb = 1 << cxt[e];
#pragma unroll
            for (int b = 0; b < DIM; ++b) {
                if ((b & cb) && !(b & tb)) {
                    const int b2 = b | tb;
                    float tr = sr[b]; sr[b] = sr[b2]; sr[b2] = tr;
                    float ti = si[b]; si[b] = si[b2]; si[b2] = ti;
                }
            }
        }
    }
#pragma unroll
    for (int b = 0; b < DIM; ++b) {
        Vr[b * DIM + col] = sr[b];
        Vi[b * DIM + col] = si[b];
    }
}

// ---------------------------------------------------------------------------
// Kernel 2: fused  angles = tanh(x @ W1[0:4]^T + b1[0:4])  via V_WMMA_F32_16X16X4_F32,
// then per-row quantum features + mu/logvar heads.
// Block = 256 threads = 8 waves. 2 row-tiles of 16 per block, K split 4-ways
// per tile (one wave each) for memory-latency hiding; partials reduced in LDS.
// ---------------------------------------------------------------------------
__global__ void __launch_bounds__(256)
qe_main(const float* __restrict__ x, const float* __restrict__ W1,
        const float* __restrict__ b1,
        const float* __restrict__ Vr, const float* __restrict__ Vi,
        const float* __restrict__ Wmu, const float* __restrict__ bmu,
        const float* __restrict__ Wlv, const float* __restrict__ blv,
        float* __restrict__ outMu, float* __restrict__ outLv)
{
    __shared__ float ldsC[8 * 256];              // 8 partial 16x16 C tiles
    __shared__ float ldsAng[ROWS_PER_BLOCK * NQ];

    const int tid  = threadIdx.x;
    const int wave = tid >> 5;
    const int lane = tid & 31;
    const int hf   = lane >> 4;    // lane half: selects K pair (A) / K row pair (B)
    const int mn   = lane & 15;    // A row index m / B column index n
    const int tile = wave >> 2;    // 0..1  -> which 16-row tile
    const int kq   = wave & 3;     // 0..3  -> which K quarter
    const int rowT = blockIdx.x * ROWS_PER_BLOCK + tile * 16;

    // f32 WMMA operand layout (ISA 7.12.2):
    //   A 16x4: lanes 0-15 hold (K,K+1) of row m; lanes 16-31 hold (K+2,K+3)
    //   B 4x16: lanes 0-15 hold rows (K,K+1) at col n; lanes 16-31 rows (K+2,K+3)
    const float* xa = x  + (size_t)(rowT + mn) * DINP + kq * KQ + 2 * hf;
    const float* wa = W1 + (size_t)(mn & 3)    * DINP + kq * KQ + 2 * hf;

    v8f acc = {0.f, 0.f, 0.f, 0.f, 0.f, 0.f, 0.f, 0.f};
    for (int k = 0; k < KQ; k += 16) {
        __builtin_prefetch(xa + k + 128, 0, 0);   // global_prefetch_b8 ahead of stream
#pragma unroll
        for (int t = 0; t < 4; ++t) {
            v2f av = *(const v2f*)(xa + k + 4 * t);
            v2f bv = *(const v2f*)(wa + k + 4 * t);
            acc = __builtin_amdgcn_wmma_f32_16x16x4_f32(
                /*neg_a=*/false, av, /*neg_b=*/false, bv,
                /*c_mod=*/(short)0, acc, /*reuse_a=*/false, /*reuse_b=*/false);
        }
    }

    // C layout: vgpr j -> (m = j + 8*hf, n = mn). Dump partials to LDS.
#pragma unroll
    for (int j = 0; j < 8; ++j)
        ldsC[wave * 256 + (j + 8 * hf) * 16 + mn] = acc[j];
    __syncthreads();

    // Reduce 4 K-partials, add bias, tanh -> angles (only n < 4 matter)
    if (tid < ROWS_PER_BLOCK * NQ) {
        const int r  = tid >> 2;       // row in block 0..31
        const int n  = tid & 3;
        const int t2 = r >> 4;
        const int m  = r & 15;
        float sum = 0.f;
#pragma unroll
        for (int u = 0; u < 4; ++u)
            sum += ldsC[(t2 * 4 + u) * 256 + m * 16 + n];
        ldsAng[tid] = tanhf(sum + b1[n]);
    }
    __syncthreads();

    // Per-row quantum simulation + output heads
    if (tid < ROWS_PER_BLOCK) {
        const int row = blockIdx.x * ROWS_PER_BLOCK + tid;
        float cq[NQ], sq[NQ];
#pragma unroll
        for (int q = 0; q < NQ; ++q) {
            const float a = ldsAng[tid * NQ + q];
            cq[q] = cosf(0.5f * a);
            sq[q] = sinf(0.5f * a);
        }
        // real product state after the RY encoding layer
        float ps[DIM];
#pragma unroll
        for (int b = 0; b < DIM; ++b) {
            float p = 1.f;
#pragma unroll
            for (int q = 0; q < NQ; ++q) p *= ((b >> q) & 1) ? sq[q] : cq[q];
            ps[b] = p;
        }
        // amp = V * ps ; probs ; signed marginals (feat i uses qubit 3-i)
        float f[NQ] = {0.f, 0.f, 0.f, 0.f};
        for (int b = 0; b < DIM; ++b) {
            float ar = 0.f, ai = 0.f;
#pragma unroll
            for (int kk = 0; kk < DIM; ++kk) {
                ar = fmaf(Vr[b * DIM + kk], ps[kk], ar);
                ai = fmaf(Vi[b * DIM + kk], ps[kk], ai);
            }
            const float p = ar * ar + ai * ai;
            f[0] += (b & 8) ? -p : p;
            f[1] += (b & 4) ? -p : p;
            f[2] += (b & 2) ? -p : p;
            f[3] += (b & 1) ? -p : p;
        }
        // mu / logvar heads (4 -> 64)
#pragma unroll 4
        for (int l = 0; l < LOUT; ++l) {
            float m  = bmu[l];
            float lv = blv[l];
#pragma unroll
            for (int i = 0; i < NQ; ++i) {
                m  = fmaf(f[i], Wmu[l * NQ + i], m);
                lv = fmaf(f[i], Wlv[l * NQ + i], lv);
            }
            outMu[(size_t)row * LOUT + l] = m;
            outLv[(size_t)row * LOUT + l] = lv;
        }
    }
}

// ---------------------------------------------------------------------------
extern "C" void kernel_launch(void* const* d_in, const int* in_sizes, int n_in,
                              void* d_out, int out_size, void* d_ws, size_t ws_size,
                              hipStream_t stream)
{
    (void)n_in; (void)out_size; (void)ws_size;
    const float* x   = (const float*)d_in[0];
    const float* W1  = (const float*)d_in[1];
    const float* b1  = (const float*)d_in[2];
    const float* qp  = (const float*)d_in[3];
    const float* Wmu = (const float*)d_in[4];
    const float* bmu = (const float*)d_in[5];
    const float* Wlv = (const float*)d_in[6];
    const float* blv = (const float*)d_in[7];
    float* out = (float*)d_out;

    const int B = in_sizes[0] / DINP;     // 32768

    float* Vr = (float*)d_ws;             // 256 floats
    float* Vi = Vr + DIM * DIM;           // 256 floats (2 KB total scratch)

    qe_build_unitary<<<1, 32, 0, stream>>>(qp, Vr, Vi);
    qe_main<<<B / ROWS_PER_BLOCK, 256, 0, stream>>>(
        x, W1, b1, Vr, Vi, Wmu, bmu, Wlv, blv,
        out, out + (size_t)B * LOUT);
}